// RMSDLoss_90804198572068
// MI455X (gfx1250) — compile-verified
//
#include <hip/hip_runtime.h>
#include <math.h>

typedef float v2f __attribute__((ext_vector_type(2)));
typedef float v8f __attribute__((ext_vector_type(8)));

#define WAVES_PER_BLOCK 2
#define NATOMS 2048

// Closed-form Kabsch RMSD from the 17 Gram sums.
// v[0:3]=Sum y', v[3:6]=Sum y, v[6:15]=Sum y'_i y_j (row major), v[15]=Sum|y'|^2, v[16]=Sum|y|^2
__device__ __forceinline__ float kabsch_rms_from_sums(const float* v, float Nf) {
    const float inv = 1.0f / Nf;
    float Sp[3] = {v[0], v[1], v[2]};
    float Sy[3] = {v[3], v[4], v[5]};
    float a[3][3];
#pragma unroll
    for (int i = 0; i < 3; ++i)
#pragma unroll
        for (int j = 0; j < 3; ++j)
            a[i][j] = v[6 + 3 * i + j] - Sp[i] * Sy[j] * inv;

    float c2 = v[15] - (Sp[0]*Sp[0] + Sp[1]*Sp[1] + Sp[2]*Sp[2]) * inv;
    float r2 = v[16] - (Sy[0]*Sy[0] + Sy[1]*Sy[1] + Sy[2]*Sy[2]) * inv;

    float det = a[0][0]*(a[1][1]*a[2][2] - a[1][2]*a[2][1])
              - a[0][1]*(a[1][0]*a[2][2] - a[1][2]*a[2][0])
              + a[0][2]*(a[1][0]*a[2][1] - a[1][1]*a[2][0]);

    // B = a^T a (symmetric PSD); singular values of a = sqrt(eig(B))
    float B00=0,B11=0,B22=0,B01=0,B02=0,B12=0;
#pragma unroll
    for (int k = 0; k < 3; ++k) {
        B00 += a[k][0]*a[k][0]; B11 += a[k][1]*a[k][1]; B22 += a[k][2]*a[k][2];
        B01 += a[k][0]*a[k][1]; B02 += a[k][0]*a[k][2]; B12 += a[k][1]*a[k][2];
    }
    // Trigonometric closed-form eigenvalues of symmetric 3x3
    float p1 = B01*B01 + B02*B02 + B12*B12;
    float q  = (B00 + B11 + B22) * (1.0f/3.0f);
    float b00 = B00 - q, b11 = B11 - q, b22 = B22 - q;
    float p2 = b00*b00 + b11*b11 + b22*b22 + 2.0f*p1;
    float p  = sqrtf(p2 * (1.0f/6.0f) + 1e-30f);
    float ip = 1.0f / p;
    float detM = b00*(b11*b22 - B12*B12)
               - B01*(B01*b22 - B12*B02)
               + B02*(B01*B12 - b11*B02);
    float rr = 0.5f * detM * ip * ip * ip;
    rr = fminf(1.0f, fmaxf(-1.0f, rr));
    float phi = acosf(rr) * (1.0f/3.0f);
    float l1 = q + 2.0f*p*cosf(phi);                       // largest
    float l3 = q + 2.0f*p*cosf(phi + 2.0943951023931953f); // smallest
    float l2 = 3.0f*q - l1 - l3;
    float s1 = sqrtf(fmaxf(l1, 0.0f));
    float s2 = sqrtf(fmaxf(l2, 0.0f));
    float s3 = sqrtf(fmaxf(l3, 0.0f));
    float s3s = (det < 0.0f) ? -s3 : s3;
    float E = c2 + r2 - 2.0f*(s1 + s2 + s3s);
    return sqrtf(fmaxf(E, 0.0f) * inv);
}

// One block = one batch pair (b0, b0+1). Each wave covers NATOMS/WAVES_PER_BLOCK atoms
// and accumulates a dual 8x8 Gram matrix X^T X with v_wmma_f32_16x16x4_f32 (4 atoms/op).
__global__ void __launch_bounds__(32 * WAVES_PER_BLOCK)
kabsch_gram_wmma(const float* __restrict__ yp, const float* __restrict__ yr,
                 float* __restrict__ rms_out) {
    const int lane = threadIdx.x & 31;
    const int wave = threadIdx.x >> 5;
    const int b0   = blockIdx.x * 2;

    const int half = lane >> 4;    // which K half of the fragment
    const int j    = lane & 15;    // matrix row/col this lane feeds
    const int bsel = j >> 3;       // 0 -> batch b0, 1 -> batch b0+1
    const int jm   = j & 7;        // role: 0-2 y', 3 const 1, 4-6 y, 7 zero
    const bool isY    = (jm >= 4) && (jm < 7);
    const bool isOne  = (jm == 3);
    const bool isZero = (jm == 7);
    const int  comp   = (jm < 3) ? jm : (isY ? (jm - 4) : 0);
    const float* base = isY ? yr : yp;

    const int atoms_per_wave = NATOMS / WAVES_PER_BLOCK;  // 1024
    const int chunks         = atoms_per_wave / 4;        // 256
    const int astart         = wave * atoms_per_wave;

    // element address for (batch, atom = astart + 2*half + v, comp); v=1 is +3 floats
    const float* p = base + ((size_t)(b0 + bsel) * NATOMS + (size_t)(astart + 2 * half)) * 3 + comp;

    v8f acc = {};  // 16x16 f32 accumulator: dual 8x8 Gram on the block diagonal

    for (int c = 0; c < chunks; c += 4) {
#pragma unroll
        for (int u = 0; u < 4; ++u) {
            float x0 = p[0];
            float x1 = p[3];
            x0 = isZero ? 0.0f : (isOne ? 1.0f : x0);
            x1 = isZero ? 0.0f : (isOne ? 1.0f : x1);
            v2f afrag = {x0, x1};
            // D = A*B + C with A == B == X-chunk fragment  ->  D += X^T X
            acc = __builtin_amdgcn_wmma_f32_16x16x4_f32(
                false, afrag, false, afrag, (short)0, acc, false, false);
            p += 12;  // advance 4 atoms * 3 comps
        }
        __builtin_prefetch(p + 96, 0, 1);  // global_prefetch ~8 chunks ahead
    }

    // Extract the 17 sums per batch half via wave shuffles.
    float ac[8];
#pragma unroll
    for (int t = 0; t < 8; ++t) ac[t] = acc[t];

    __shared__ float part[WAVES_PER_BLOCK][2][17];

    float vals[2][17];
#pragma unroll
    for (int h = 0; h < 2; ++h) {
        const int lb = 24 * h;  // lane base for block-diagonal (rows 8h+m live in lanes 16h+N)
#pragma unroll
        for (int i = 0; i < 3; ++i)
            vals[h][i] = __shfl(ac[i], lb + 3, 32);            // Sum y'_i  (col "1")
#pragma unroll
        for (int jj = 0; jj < 3; ++jj)
            vals[h][3 + jj] = __shfl(ac[3], lb + 4 + jj, 32);  // Sum y_j   (row "1")
#pragma unroll
        for (int i = 0; i < 3; ++i)
#pragma unroll
            for (int jj = 0; jj < 3; ++jj)
                vals[h][6 + 3 * i + jj] = __shfl(ac[i], lb + 4 + jj, 32);  // Sum y'_i y_j
        vals[h][15] = __shfl(ac[0], lb + 0, 32) + __shfl(ac[1], lb + 1, 32)
                    + __shfl(ac[2], lb + 2, 32);               // Sum |y'|^2
        vals[h][16] = __shfl(ac[4], lb + 4, 32) + __shfl(ac[5], lb + 5, 32)
                    + __shfl(ac[6], lb + 6, 32);               // Sum |y|^2
    }

    if (lane == 0) {
#pragma unroll
        for (int h = 0; h < 2; ++h)
#pragma unroll
            for (int k = 0; k < 17; ++k) part[wave][h][k] = vals[h][k];
    }
    __syncthreads();

    if (threadIdx.x == 0) {
#pragma unroll
        for (int h = 0; h < 2; ++h) {
            float v[17];
#pragma unroll
            for (int k = 0; k < 17; ++k) {
                float s = 0.0f;
#pragma unroll
                for (int w = 0; w < WAVES_PER_BLOCK; ++w) s += part[w][h][k];
                v[k] = s;
            }
            rms_out[b0 + h] = kabsch_rms_from_sums(v, (float)NATOMS);
        }
    }
}

__global__ void __launch_bounds__(256)
reduce_mean_kernel(const float* __restrict__ rms, float* __restrict__ out, int n) {
    __shared__ float s[256];
    int t = threadIdx.x;
    float v = 0.0f;
    for (int i = t; i < n; i += 256) v += rms[i];
    s[t] = v;
    __syncthreads();
    for (int k = 128; k > 0; k >>= 1) {
        if (t < k) s[t] += s[t + k];
        __syncthreads();
    }
    if (t == 0) out[0] = s[0] / (float)n;
}

extern "C" void kernel_launch(void* const* d_in, const int* in_sizes, int n_in,
                              void* d_out, int out_size, void* d_ws, size_t ws_size,
                              hipStream_t stream) {
    const float* y_prime = (const float*)d_in[0];
    const float* y_ref   = (const float*)d_in[1];
    float* out = (float*)d_out;
    float* rms = (float*)d_ws;  // B floats of scratch

    const long total = (long)in_sizes[0];
    const int  B = (int)(total / (3L * NATOMS));  // 2048

    kabsch_gram_wmma<<<dim3(B / 2), dim3(32 * WAVES_PER_BLOCK), 0, stream>>>(y_prime, y_ref, rms);
    reduce_mean_kernel<<<dim3(1), dim3(256), 0, stream>>>(rms, out, B);
}